// gnn_based_seq_2302102471103
// MI455X (gfx1250) — compile-verified
//
#include <hip/hip_runtime.h>
#include <hip/hip_bf16.h>

#define NNODES   50000
#define NEDGES   1600000
#define SEQDIM   512
#define HDIM     128
#define LN_EPS   1e-5f

typedef float v2f __attribute__((ext_vector_type(2)));
typedef float v8f __attribute__((ext_vector_type(8)));

// ---------------------------------------------------------------------------
// Degree helpers
// ---------------------------------------------------------------------------
__global__ void k_zero_f32(float* __restrict__ p, int n) {
    int i = blockIdx.x * blockDim.x + threadIdx.x;
    if (i < n) p[i] = 0.0f;
}

__global__ void k_deg_count(const int* __restrict__ dst, float* __restrict__ deg, int e) {
    int i = blockIdx.x * blockDim.x + threadIdx.x;
    if (i < e) atomicAdd(&deg[dst[i]], 1.0f);
}

__global__ void k_deg_to_dinv(float* __restrict__ deg, int n) {
    int i = blockIdx.x * blockDim.x + threadIdx.x;
    if (i < n) deg[i] = rsqrtf(deg[i] + 1.0f);   // +1 = self loop
}

// ---------------------------------------------------------------------------
// fp32 WMMA GEMM: out[n,128] = act(A[n,K] @ W[K,128])
// Block = 256 threads = 8 wave32; block computes a 16-row x 128-col tile,
// wave w owns a 16x16 output tile (cols 16w..16w+15).
// Uses V_WMMA_F32_16X16X4_F32: exact fp32 math, matrix pipe.
// ---------------------------------------------------------------------------
__global__ __launch_bounds__(256)
void k_gemm_wmma_f32(const float* __restrict__ A, const float* __restrict__ W,
                     float* __restrict__ out, int K, int relu) {
    const int wave = threadIdx.x >> 5;
    const int lane = threadIdx.x & 31;
    const int row0 = blockIdx.x << 4;      // 16 rows per block (N % 16 == 0)
    const int col0 = wave << 4;            // 16 cols per wave
    const int m    = lane & 15;            // A row within tile
    const int n    = lane & 15;            // B/D col within tile
    const int kh   = lane >> 4;            // half-wave: K pair {0,1} vs {2,3}

    v8f acc = {};
    const float* __restrict__ arow = A + (size_t)(row0 + m) * K;
    const float* __restrict__ wcol = W + col0 + n;

    for (int k = 0; k < K; k += 4) {
        // A fragment: this lane holds A[row0+m][k+2*kh .. k+2*kh+1]
        v2f a = *(const v2f*)(arow + k + 2 * kh);
        // B fragment: this lane holds W[k+2*kh][col0+n], W[k+2*kh+1][col0+n]
        v2f b;
        b.x = wcol[(size_t)(k + 2 * kh) * HDIM];
        b.y = wcol[(size_t)(k + 2 * kh + 1) * HDIM];
        acc = __builtin_amdgcn_wmma_f32_16x16x4_f32(
                  false, a, false, b, (short)0, acc, false, false);
    }

    // D layout: lanes 0-15 -> M = v, lanes 16-31 -> M = v + 8; N = lane & 15
    const int mbase = kh << 3;
#pragma unroll
    for (int v = 0; v < 8; ++v) {
        float r = acc[v];
        if (relu) r = fmaxf(r, 0.0f);
        out[(size_t)(row0 + mbase + v) * HDIM + col0 + n] = r;
    }
}

// ---------------------------------------------------------------------------
// Self-loop init + bias: agg[i][c] = h[i][c] * dinv[i]^2 + b[c]
// ---------------------------------------------------------------------------
__global__ void k_self_bias(const float* __restrict__ h, const float* __restrict__ dinv,
                            const float* __restrict__ bias, float* __restrict__ agg) {
    int i = blockIdx.x * blockDim.x + threadIdx.x;
    if (i >= NNODES * HDIM) return;
    int r = i >> 7, c = i & 127;
    float di = dinv[r];
    agg[i] = h[i] * di * di + bias[c];
}

// ---------------------------------------------------------------------------
// Edge scatter: one wave32 per edge; each lane handles 4 contiguous features.
// agg[dst] += h[src] * dinv[src] * dinv[dst]
// ---------------------------------------------------------------------------
__global__ __launch_bounds__(256)
void k_edge_scatter(const int* __restrict__ src, const int* __restrict__ dst,
                    const float* __restrict__ h, const float* __restrict__ dinv,
                    float* __restrict__ agg, int e) {
    const int wave = threadIdx.x >> 5;
    const int lane = threadIdx.x & 31;
    const int edge = (blockIdx.x << 3) + wave;   // 8 edges per block
    if (edge >= e) return;

    const int s = src[edge];
    const int d = dst[edge];
    const float norm = dinv[s] * dinv[d];

    const float4 hv = *(const float4*)(h + (size_t)s * HDIM + (lane << 2));
    float* ap = agg + (size_t)d * HDIM + (lane << 2);
    atomicAdd(ap + 0, hv.x * norm);
    atomicAdd(ap + 1, hv.y * norm);
    atomicAdd(ap + 2, hv.z * norm);
    atomicAdd(ap + 3, hv.w * norm);
}

// ---------------------------------------------------------------------------
// Fused add + LayerNorm (+optional relu): out = act(LN(agg + x2 + 1e-6))
// One wave32 per row (128 feats = 4/lane), wave32 shuffle reductions.
// ---------------------------------------------------------------------------
__global__ __launch_bounds__(256)
void k_add_ln(const float* __restrict__ agg, const float* __restrict__ x2,
              const float* __restrict__ g, const float* __restrict__ b,
              float* __restrict__ out, int relu) {
    const int wave = threadIdx.x >> 5;
    const int lane = threadIdx.x & 31;
    const int row  = (blockIdx.x << 3) + wave;
    if (row >= NNODES) return;

    const float4 av = *(const float4*)(agg + (size_t)row * HDIM + (lane << 2));
    const float4 xv = *(const float4*)(x2  + (size_t)row * HDIM + (lane << 2));
    float v0 = av.x + xv.x + 1e-6f;
    float v1 = av.y + xv.y + 1e-6f;
    float v2 = av.z + xv.z + 1e-6f;
    float v3 = av.w + xv.w + 1e-6f;

    float s = v0 + v1 + v2 + v3;
#pragma unroll
    for (int off = 16; off > 0; off >>= 1) s += __shfl_xor(s, off, 32);
    const float mu = s * (1.0f / 128.0f);

    float d0 = v0 - mu, d1 = v1 - mu, d2 = v2 - mu, d3 = v3 - mu;
    float q = d0 * d0 + d1 * d1 + d2 * d2 + d3 * d3;
#pragma unroll
    for (int off = 16; off > 0; off >>= 1) q += __shfl_xor(q, off, 32);
    const float inv = rsqrtf(q * (1.0f / 128.0f) + LN_EPS);

    const int c = lane << 2;
    float4 r;
    r.x = d0 * inv * g[c + 0] + b[c + 0];
    r.y = d1 * inv * g[c + 1] + b[c + 1];
    r.z = d2 * inv * g[c + 2] + b[c + 2];
    r.w = d3 * inv * g[c + 3] + b[c + 3];
    if (relu) {
        r.x = fmaxf(r.x, 0.0f); r.y = fmaxf(r.y, 0.0f);
        r.z = fmaxf(r.z, 0.0f); r.w = fmaxf(r.w, 0.0f);
    }
    *(float4*)(out + (size_t)row * HDIM + c) = r;
}

// ---------------------------------------------------------------------------
extern "C" void kernel_launch(void* const* d_in, const int* in_sizes, int n_in,
                              void* d_out, int out_size, void* d_ws, size_t ws_size,
                              hipStream_t stream) {
    (void)in_sizes; (void)n_in; (void)out_size; (void)ws_size;

    const int*   mat    = (const int*)d_in[0];       // [2,E] int32
    const int*   src    = mat;
    const int*   dst    = mat + NEDGES;
    const float* seq    = (const float*)d_in[1];
    const float* W_seq  = (const float*)d_in[2];
    const float* gcn1_W = (const float*)d_in[3];
    const float* gcn1_b = (const float*)d_in[4];
    const float* lin1_W = (const float*)d_in[5];
    const float* ln1_g  = (const float*)d_in[6];
    const float* ln1_b  = (const float*)d_in[7];
    const float* gcn2_W = (const float*)d_in[8];
    const float* gcn2_b = (const float*)d_in[9];
    const float* lin2_W = (const float*)d_in[10];
    const float* ln2_g  = (const float*)d_in[11];
    const float* ln2_b  = (const float*)d_in[12];

    float* out    = (float*)d_out;
    float* x_seq  = out;                               // output 0: [N,128]
    float* x_fin  = out + (size_t)NNODES * HDIM;       // output 1 (also x1 staging)

    float* ws   = (float*)d_ws;
    float* dinv = ws;                                  // [N]
    float* h    = ws + NNODES;                         // [N,128]
    float* x2   = h  + (size_t)NNODES * HDIM;          // [N,128]
    float* agg  = x2 + (size_t)NNODES * HDIM;          // [N,128]

    const int nh = NNODES * HDIM;
    const dim3 blk(256);

    // --- degrees -> dinv (shared by both layers) ---
    k_zero_f32  <<<(NNODES + 255) / 256, blk, 0, stream>>>(dinv, NNODES);
    k_deg_count <<<(NEDGES + 255) / 256, blk, 0, stream>>>(dst, dinv, NEDGES);
    k_deg_to_dinv<<<(NNODES + 255) / 256, blk, 0, stream>>>(dinv, NNODES);

    // --- x_seq = relu(seq @ W_seq) ---
    k_gemm_wmma_f32<<<NNODES / 16, blk, 0, stream>>>(seq, W_seq, x_seq, SEQDIM, 1);

    // --- layer 1 ---
    k_gemm_wmma_f32<<<NNODES / 16, blk, 0, stream>>>(x_seq, gcn1_W, h,  HDIM, 0);
    k_gemm_wmma_f32<<<NNODES / 16, blk, 0, stream>>>(x_seq, lin1_W, x2, HDIM, 0);
    k_self_bias    <<<(nh + 255) / 256, blk, 0, stream>>>(h, dinv, gcn1_b, agg);
    k_edge_scatter <<<(NEDGES + 7) / 8, blk, 0, stream>>>(src, dst, h, dinv, agg, NEDGES);
    k_add_ln       <<<(NNODES + 7) / 8, blk, 0, stream>>>(agg, x2, ln1_g, ln1_b, x_fin, 1);

    // --- layer 2 (reads x1 from x_fin, final LN overwrites x_fin) ---
    k_gemm_wmma_f32<<<NNODES / 16, blk, 0, stream>>>(x_fin, gcn2_W, h,  HDIM, 0);
    k_gemm_wmma_f32<<<NNODES / 16, blk, 0, stream>>>(x_fin, lin2_W, x2, HDIM, 0);
    k_self_bias    <<<(nh + 255) / 256, blk, 0, stream>>>(h, dinv, gcn2_b, agg);
    k_edge_scatter <<<(NEDGES + 7) / 8, blk, 0, stream>>>(src, dst, h, dinv, agg, NEDGES);
    k_add_ln       <<<(NNODES + 7) / 8, blk, 0, stream>>>(agg, x2, ln2_g, ln2_b, x_fin, 0);
}